// MyGCN_86706799772241
// MI455X (gfx1250) — compile-verified
//
#include <hip/hip_runtime.h>

#define HID 128
#define EPSV 1e-5f

typedef __attribute__((ext_vector_type(2))) float v2f;
typedef __attribute__((ext_vector_type(8))) float v8f;

__device__ __forceinline__ void atomAddF(float* p, float v) {
    unsafeAtomicAdd(p, v);   // global_atomic_add_f32 (no CAS loop)
}

// ---------------- degree / edge-weight normalization ----------------
__global__ __launch_bounds__(256) void k_deg(const int* __restrict__ row,
                                             const float* __restrict__ ew,
                                             float* __restrict__ deg, int nE) {
    int e = blockIdx.x * 256 + threadIdx.x;
    if (e >= nE) return;
    atomAddF(&deg[row[e]], ew[e]);
}

__global__ __launch_bounds__(256) void k_invdeg(float* __restrict__ deg, int n) {
    int i = blockIdx.x * 256 + threadIdx.x;
    if (i >= n) return;
    float d = deg[i];
    d = (d < 0.5f) ? d + 1.0f : d;
    deg[i] = 1.0f / d;
}

__global__ __launch_bounds__(256) void k_wnorm(const int* __restrict__ row,
                                               const float* __restrict__ ew,
                                               const float* __restrict__ invdeg,
                                               float* __restrict__ wn, int nE) {
    int e = blockIdx.x * 256 + threadIdx.x;
    if (e >= nE) return;
    wn[e] = ew[e] * invdeg[row[e]];
}

// ---------------- embedding gather: x = emb[x_idx] ----------------
__global__ __launch_bounds__(256) void k_gather(const int* __restrict__ idx,
                                                const float* __restrict__ emb,
                                                float* __restrict__ x, int n) {
    int t = blockIdx.x * 256 + threadIdx.x;
    int node = t >> 5;
    int lane = t & 31;
    if (node >= n) return;
    const float4* src = (const float4*)(emb + (size_t)idx[node] * HID) + lane;
    float4* dst = (float4*)(x + (size_t)node * HID) + lane;
    *dst = *src;
}

// ---------------- WMMA fp32 GEMM: out = act(A[N,128]@W[128,128] (+bias) (+Cin)) ----------------
__global__ __launch_bounds__(256) void k_gemm(const float* __restrict__ A,
                                              const float* __restrict__ W,
                                              const float* __restrict__ bias,
                                              const float* Cin,
                                              float* __restrict__ out,
                                              int nrows, int relu) {
    __shared__ float wl[HID * HID];

    // ---- stage 64 KB of weights into LDS via CDNA5 async copy (ASYNCcnt path,
    //      no VGPR round-trip). GVS addressing: vdst = LDS byte offset,
    //      vaddr = 32-bit byte offset, saddr = uniform 64-bit base.
    {
        unsigned ldsW = (unsigned)(size_t)wl;        // low 32 bits of flat LDS ptr = LDS offset
        unsigned base = threadIdx.x * 16u;           // 16 B per lane per op
        #pragma unroll
        for (int i = 0; i < 16; ++i) {               // 256 thr * 16 it * 16 B = 64 KB
            unsigned go = base + (unsigned)i * 4096u;
            asm volatile("global_load_async_to_lds_b128 %0, %1, %2"
                         :: "v"(ldsW + go), "v"(go), "s"(W)
                         : "memory");
        }
        asm volatile("s_wait_asynccnt 0x0" ::: "memory");
    }
    __syncthreads();

    const int tid  = threadIdx.x;
    const int wave = tid >> 5;
    const int lane = tid & 31;
    const int row0 = blockIdx.x * 128 + wave * 16;
    if (row0 >= nrows) return;              // wave-uniform: EXEC stays all-1s

    const int mrow  = lane & 15;            // A row / B,C column within tile
    const int khalf = (lane >> 4) * 2;      // K sub-pair 0 or 2
    const int roff  = (lane >> 4) * 8;      // C row offset for upper lanes

    v8f c[8];
    if (Cin) {
        #pragma unroll
        for (int t = 0; t < 8; ++t)
            #pragma unroll
            for (int v = 0; v < 8; ++v)
                c[t][v] = Cin[(size_t)(row0 + roff + v) * HID + t * 16 + mrow];
    } else {
        #pragma unroll
        for (int t = 0; t < 8; ++t) {
            v8f z = {0.f, 0.f, 0.f, 0.f, 0.f, 0.f, 0.f, 0.f};
            c[t] = z;
        }
    }

    // software-pipelined A-fragment loads: next k-step's load is in flight
    // while the current 8 WMMAs execute.
    const float* arow = A + (size_t)(row0 + mrow) * HID + khalf;
    v2f a_cur = *(const v2f*)(arow);
    #pragma unroll
    for (int k0 = 0; k0 < HID; k0 += 4) {
        v2f a_nxt = a_cur;
        if (k0 + 4 < HID) a_nxt = *(const v2f*)(arow + k0 + 4);
        #pragma unroll
        for (int t = 0; t < 8; ++t) {
            v2f b;
            b.x = wl[(k0 + khalf + 0) * HID + t * 16 + mrow];
            b.y = wl[(k0 + khalf + 1) * HID + t * 16 + mrow];
            c[t] = __builtin_amdgcn_wmma_f32_16x16x4_f32(
                       false, a_cur, false, b, (short)0, c[t], false, false);
        }
        a_cur = a_nxt;
    }

    #pragma unroll
    for (int t = 0; t < 8; ++t) {
        int col0 = t * 16 + mrow;
        float bv = bias ? bias[col0] : 0.0f;
        #pragma unroll
        for (int v = 0; v < 8; ++v) {
            float val = c[t][v] + bv;
            if (relu) val = fmaxf(val, 0.0f);
            out[(size_t)(row0 + roff + v) * HID + col0] = val;
        }
    }
}

// ---------------- edge scatter: m[row] += wn * h[col]  (L2-resident atomics) ----------------
__global__ __launch_bounds__(256) void k_scatter(const int* __restrict__ row,
                                                 const int* __restrict__ col,
                                                 const float* __restrict__ wn,
                                                 const float* __restrict__ h,
                                                 float* __restrict__ m, int nE) {
    int t = blockIdx.x * 256 + threadIdx.x;
    int e = t >> 5;
    int lane = t & 31;
    if (e >= nE) return;
    int r = row[e], cc = col[e];
    float w = wn[e];
    const float4 hv = *((const float4*)(h + (size_t)cc * HID) + lane);
    float* dst = m + (size_t)r * HID + lane * 4;
    atomAddF(dst + 0, w * hv.x);
    atomAddF(dst + 1, w * hv.y);
    atomAddF(dst + 2, w * hv.z);
    atomAddF(dst + 3, w * hv.w);
}

// ---------------- per-feature sum / sumsq over nodes ----------------
__global__ __launch_bounds__(256) void k_stats(const float* __restrict__ x, int n,
                                               float* __restrict__ stats) {
    int f = threadIdx.x & 127;
    int half = threadIdx.x >> 7;
    float s = 0.f, s2 = 0.f;
    for (int nd = blockIdx.x * 2 + half; nd < n; nd += gridDim.x * 2) {
        float v = x[(size_t)nd * HID + f];
        s += v;
        s2 += v * v;
    }
    atomAddF(&stats[f], s);
    atomAddF(&stats[HID + f], s2);
}

// ---------------- GraphNorm (+ optional ReLU), in place ----------------
__global__ __launch_bounds__(256) void k_gnorm(float* __restrict__ x,
                                               const float* __restrict__ stats,
                                               const float* __restrict__ g,
                                               const float* __restrict__ b,
                                               const float* __restrict__ a,
                                               int n, int relu) {
    int t = blockIdx.x * 256 + threadIdx.x;
    if (t >= n * HID) return;
    int f = t & 127;
    float invN = 1.0f / (float)n;
    float mean = stats[f] * invN;
    float ex2  = stats[HID + f] * invN;
    float al = a[f];
    float var = ex2 - (2.0f * al - al * al) * mean * mean;
    float scale = g[f] * rsqrtf(var + EPSV);
    float v = (x[t] - al * mean) * scale + b[f];
    if (relu) v = fmaxf(v, 0.0f);
    x[t] = v;
}

// =====================================================================
extern "C" void kernel_launch(void* const* d_in, const int* in_sizes, int n_in,
                              void* d_out, int out_size, void* d_ws, size_t ws_size,
                              hipStream_t stream) {
    const int*   x_idx = (const int*)d_in[0];
    const int*   eidx  = (const int*)d_in[1];
    const float* ew    = (const float*)d_in[2];
    const float* emb   = (const float*)d_in[3];
    const float* Wt    = (const float*)d_in[4];
    const float* bt    = (const float*)d_in[5];
    const float* Wc    = (const float*)d_in[6];
    const float* bc    = (const float*)d_in[7];
    const float* cg_g  = (const float*)d_in[8];
    const float* cg_b  = (const float*)d_in[9];
    const float* cg_a  = (const float*)d_in[10];
    const float* gn_g  = (const float*)d_in[11];
    const float* gn_b  = (const float*)d_in[12];
    const float* gn_a  = (const float*)d_in[13];

    const int N = in_sizes[0];
    const int E = in_sizes[2];
    const int* row = eidx;
    const int* col = eidx + E;

    // workspace carve
    char* ws = (char*)d_ws;
    size_t off = 0;
    auto carve = [&](size_t bytes) {
        void* p = ws + off;
        off += (bytes + 255) & ~(size_t)255;
        return p;
    };
    const size_t featB = (size_t)N * HID * sizeof(float);
    float* xA    = (float*)carve(featB);
    float* xB    = (float*)carve(featB);
    float* h     = (float*)carve(featB);
    float* m     = (float*)carve(featB);
    float* wn    = (float*)carve((size_t)E * sizeof(float));
    float* deg   = (float*)carve((size_t)N * sizeof(float));
    float* stats = (float*)carve(2 * HID * sizeof(float));

    dim3 blk(256);
    int eBlocks    = (E + 255) / 256;
    int nBlocks    = (N + 255) / 256;
    int gemmBlocks = (N + 127) / 128;
    int gathBlocks = (N * 32 + 255) / 256;
    int scatBlocks = (int)(((long long)E * 32 + 255) / 256);
    int elemBlocks = (int)(((long long)N * HID + 255) / 256);

    // mean-aggregation weights
    hipMemsetAsync(deg, 0, (size_t)N * sizeof(float), stream);
    k_deg<<<eBlocks, blk, 0, stream>>>(row, ew, deg, E);
    k_invdeg<<<nBlocks, blk, 0, stream>>>(deg, N);
    k_wnorm<<<eBlocks, blk, 0, stream>>>(row, ew, deg, wn, E);

    // x = emb[x_idx]
    k_gather<<<gathBlocks, blk, 0, stream>>>(x_idx, emb, xA, N);

    const float* xcur = xA;
    float* xnext = xB;
    for (int i = 0; i < 3; ++i) {
        const float* Wti = Wt + (size_t)i * HID * HID;
        const float* Wc1 = Wc + (size_t)i * 2 * HID * HID;      // rows 0..127  (m part)
        const float* Wc2 = Wc1 + (size_t)HID * HID;             // rows 128..255 (x part)

        // h = relu(x @ Wt + bt)
        k_gemm<<<gemmBlocks, blk, 0, stream>>>(xcur, Wti, bt + i * HID, nullptr, h, N, 1);

        // m = segment_sum(wn * h[col], row)
        hipMemsetAsync(m, 0, featB, stream);
        k_scatter<<<scatBlocks, blk, 0, stream>>>(row, col, wn, h, m, E);

        // m = graph_norm(m, cg params)
        hipMemsetAsync(stats, 0, 2 * HID * sizeof(float), stream);
        k_stats<<<256, blk, 0, stream>>>(m, N, stats);
        k_gnorm<<<elemBlocks, blk, 0, stream>>>(m, stats, cg_g + i * HID, cg_b + i * HID,
                                                cg_a + i * HID, N, 0);

        // out = concat([m, x]) @ Wc + bc   (split into two GEMMs)
        float* dst = (i == 2) ? (float*)d_out : xnext;
        k_gemm<<<gemmBlocks, blk, 0, stream>>>(m, Wc1, bc + i * HID, nullptr, dst, N, 0);
        k_gemm<<<gemmBlocks, blk, 0, stream>>>(xcur, Wc2, nullptr, dst, dst, N, 0);

        if (i < 2) {
            hipMemsetAsync(stats, 0, 2 * HID * sizeof(float), stream);
            k_stats<<<256, blk, 0, stream>>>(dst, N, stats);
            k_gnorm<<<elemBlocks, blk, 0, stream>>>(dst, stats, gn_g + i * HID,
                                                    gn_b + i * HID, gn_a + i * HID, N, 1);
            xcur = dst;
            xnext = (dst == xB) ? xA : xB;
        }
    }
}